// Diffusion_31044023615893
// MI455X (gfx1250) — compile-verified
//
#include <hip/hip_runtime.h>
#include <hip/hip_bf16.h>
#include <math.h>

// ---------------------------------------------------------------------------
// Problem constants (from reference)
// ---------------------------------------------------------------------------
constexpr int BS   = 128;
constexpr int NS   = 1024;
constexpr int DIM  = 6;
constexpr int CTX  = 128;
constexpr int HID  = 512;
constexpr int TT   = 100;
constexpr int ITERS = 8;

// workspace layout (float slots)
constexpr size_t OFF_SA    = 0;                       // sqrt_acp[100]   (pad 128)
constexpr size_t OFF_S1    = 128;                     // sqrt_1m_acp[100](pad 128)
constexpr size_t OFF_CTR   = 256;                     // int counts[8], cont[8], done
constexpr size_t OFF_BIAS  = 512;                     // bias[BS*HID]
constexpr size_t OFF_NOISE = OFF_BIAS  + (size_t)BS*HID;      // 66048
constexpr size_t OFF_NOISY = OFF_NOISE + (size_t)BS*NS*DIM;   // 852480
constexpr size_t OFF_MISM  = OFF_NOISY + (size_t)BS*NS*DIM;   // 1638912 (ints)

typedef float v2f __attribute__((ext_vector_type(2)));
typedef float v8f __attribute__((ext_vector_type(8)));

__device__ __forceinline__ v8f wmma4(v2f a, v2f b, v8f c) {
  // D(16x16,f32) = A(16x4,f32) * B(4x16,f32) + C
  return __builtin_amdgcn_wmma_f32_16x16x4_f32(false, a, false, b, (short)0, c,
                                               false, false);
}

__device__ __forceinline__ unsigned hash_u32(unsigned x) {
  x ^= x >> 16; x *= 0x7feb352dU;
  x ^= x >> 15; x *= 0x846ca68bU;
  x ^= x >> 16;
  return x;
}

// monotonic float -> uint key (total order matches float order for all finite)
__device__ __forceinline__ unsigned fkey(float f) {
  unsigned b = __float_as_uint(f);
  return b ^ ((unsigned)((int)b >> 31) | 0x80000000u);
}

// ---------------------------------------------------------------------------
// 0) scheduler tables (squaredcos_cap_v2 cumprod) + zero counters. 1 thread.
// ---------------------------------------------------------------------------
__global__ void tables_kernel(float* __restrict__ wsf) {
  if (threadIdx.x != 0 || blockIdx.x != 0) return;
  const double PI = 3.14159265358979323846;
  double acp = 1.0;
  for (int i = 0; i < TT; ++i) {
    double t0 = (double)i / TT, t1 = (double)(i + 1) / TT;
    double ab0 = cos((t0 + 0.008) / 1.008 * PI * 0.5); ab0 *= ab0;
    double ab1 = cos((t1 + 0.008) / 1.008 * PI * 0.5); ab1 *= ab1;
    double beta = 1.0 - ab1 / ab0;
    if (beta > 0.999) beta = 0.999;
    acp *= (1.0 - beta);
    wsf[OFF_SA + i] = (float)sqrt(acp);
    wsf[OFF_S1 + i] = (float)sqrt(1.0 - acp);
  }
  int* ctr = (int*)(wsf + OFF_CTR);
  for (int i = 0; i < 17; ++i) ctr[i] = 0;
}

// ---------------------------------------------------------------------------
// 1) prep: noise state = noise0 ; noisy = sqrt_acp[t]*data + sqrt_1m_acp[t]*noise0
// ---------------------------------------------------------------------------
__global__ void prep_kernel(const float* __restrict__ data,
                            const float* __restrict__ noise0,
                            const int* __restrict__ timesteps,
                            float* __restrict__ wsf) {
  const size_t gid = (size_t)blockIdx.x * blockDim.x + threadIdx.x;
  if (gid >= (size_t)BS * NS * DIM) return;
  const int b = (int)(gid / (NS * DIM));
  const int t = timesteps[b];
  const float sa = wsf[OFF_SA + t], s1 = wsf[OFF_S1 + t];
  const float n0 = noise0[gid];
  wsf[OFF_NOISE + gid] = n0;
  wsf[OFF_NOISY + gid] = sa * data[gid] + s1 * n0;
}

// ---------------------------------------------------------------------------
// 2) per-batch hidden bias: context@Wc + (t/T)*Wt + b1
// ---------------------------------------------------------------------------
__global__ void bias_kernel(const float* __restrict__ context,
                            const float* __restrict__ Wc,
                            const float* __restrict__ Wt,
                            const float* __restrict__ b1,
                            const int* __restrict__ timesteps,
                            float* __restrict__ wsf) {
  const int b = blockIdx.x;
  const float temb = (float)timesteps[b] / (float)TT;
  const float* cb = context + (size_t)b * CTX;
  for (int h = threadIdx.x; h < HID; h += blockDim.x) {
    float s = 0.f;
    for (int c = 0; c < CTX; ++c) s += cb[c] * Wc[c * HID + h];
    wsf[OFF_BIAS + (size_t)b * HID + h] = s + temb * Wt[h] + b1[h];
  }
}

// ---------------------------------------------------------------------------
// 3) WMMA pairwise-distance argmin.
//    score[m][n] = d_sq[m] - 2 * dot(data[m], nd[n]); argmin_m per n.
//    A = data tile (M=16 x K), B = -2*nd tile (K x N=16), C init = d_sq[m].
//    grid = BS*8 blocks x 256 threads (8 waves; wave = one 16-wide n tile).
// ---------------------------------------------------------------------------
__global__ void argmin_kernel(const float* __restrict__ data,
                              float* __restrict__ wsf, int it) {
  const int b    = blockIdx.x >> 3;
  const int ngrp = blockIdx.x & 7;
  const int tid  = threadIdx.x;
  const int wave = tid >> 5;
  const int lane = tid & 31;
  const bool hi  = lane >= 16;
  const int  lm  = lane & 15;

  __shared__ __align__(32) float sD[NS * DIM];   // data[b] : 24 KB
  __shared__ __align__(32) float sDsq[NS];       // row norms: 4 KB

  const float* db = data + (size_t)b * NS * DIM;

  // stage data[b] into LDS with CDNA5 async copies (no VGPR round-trip).
  // GVS addressing: mem = SADDR(64b) + VADDR(32b offset); VDST = LDS byte addr.
  for (int i = tid * 4; i < NS * DIM; i += 256 * 4) {
    unsigned lds_addr = (unsigned)(uintptr_t)&sD[i];
    unsigned voff     = (unsigned)(i * 4);
    asm volatile("global_load_async_to_lds_b128 %0, %1, %2"
                 :: "v"(lds_addr), "v"(voff), "s"(db) : "memory");
  }
  asm volatile("s_wait_asynccnt 0x0" ::: "memory");
  __syncthreads();

  for (int m = tid; m < NS; m += 256) {
    float s = 0.f;
    #pragma unroll
    for (int k = 0; k < DIM; ++k) { float v = sD[m * DIM + k]; s += v * v; }
    sDsq[m] = s;
  }
  __syncthreads();

  const int n0 = (ngrp * 8 + wave) * 16;
  const int nl = n0 + lm;

  // nd[n] = {data[n][0..2], noisy[n][3..5]}, scaled by -2 into B operand.
  // Layout per lane (B 4x16, K x N): lanes 0-15 K={0,1}, lanes 16-31 K={2,3};
  // second WMMA covers K'={4,5} (lanes 0-15) with zero padding on hi lanes.
  const float* noisyB = wsf + OFF_NOISY + ((size_t)b * NS + nl) * DIM;
  v2f b0, b1v;
  {
    const float x0 = sD[nl * DIM + (hi ? 2 : 0)];
    const float x1 = hi ? noisyB[3] : sD[nl * DIM + 1];
    b0[0] = -2.f * x0;
    b0[1] = -2.f * x1;
    const float y0 = noisyB[4], y1 = noisyB[5];  // unconditional, valid addr
    b1v[0] = hi ? 0.f : -2.f * y0;
    b1v[1] = hi ? 0.f : -2.f * y1;
  }

  // per-lane element offsets for A-tile pair loads (even -> 8B aligned)
  const int aoff0 = lm * DIM + (hi ? 2 : 0);
  const int aoff1 = lm * DIM + 4;
  const int rbase = hi ? 8 : 0;

  // packed (scoreKey << 32 | m) running minimum: branchless, ties -> lowest m
  unsigned long long best = 0xFFFFFFFFFFFFFFFFull;

  for (int mt = 0; mt < NS / 16; ++mt) {
    const int m0 = mt * 16;
    // branchless A loads: all lanes issue the same ds_load_b64s
    const float2 pa0 = *(const float2*)&sD[m0 * DIM + aoff0];
    const float2 pa1 = *(const float2*)&sD[m0 * DIM + aoff1];
    v2f a0, a1;
    a0[0] = pa0.x;            a0[1] = pa0.y;
    a1[0] = hi ? 0.f : pa1.x; a1[1] = hi ? 0.f : pa1.y;   // v_cndmask x2

    const int mrow = m0 + rbase;
    const float4 q0 = *(const float4*)&sDsq[mrow];
    const float4 q1 = *(const float4*)&sDsq[mrow + 4];
    v8f c = {q0.x, q0.y, q0.z, q0.w, q1.x, q1.y, q1.z, q1.w};

    c = wmma4(a0, b0, c);
    c = wmma4(a1, b1v, c);

    #pragma unroll
    for (int v = 0; v < 8; ++v) {
      const unsigned long long key =
          ((unsigned long long)fkey(c[v]) << 32) | (unsigned)(mrow + v);
      best = key < best ? key : best;   // v_min_u64
    }
  }

  // merge the two wave halves (rows m+0..7 vs m+8..15 of each tile)
  const unsigned long long other = __shfl_xor(best, 16, 32);
  best = other < best ? other : best;
  const int bestIdx = (int)(unsigned)(best & 0xFFFFFFFFull);

  int* mismB = (int*)(wsf + OFF_MISM);
  int* ctr   = (int*)(wsf + OFF_CTR);
  if (lane < 16) {
    const int mism = (bestIdx != nl) ? 1 : 0;
    mismB[b * NS + nl] = mism;
    const unsigned long long bal = __ballot(mism);
    if (lane == 0) atomicAdd(&ctr[it], (int)__popcll(bal));
  }
}

// ---------------------------------------------------------------------------
// 4) decide: cont = (count >= 10) && !done ; done |= !cont
// ---------------------------------------------------------------------------
__global__ void decide_kernel(float* __restrict__ wsf, int it) {
  if (threadIdx.x || blockIdx.x) return;
  int* ctr = (int*)(wsf + OFF_CTR);
  const int done = ctr[16];
  const int cont = (ctr[it] >= 10) && !done;
  ctr[8 + it] = cont;
  if (!cont) ctr[16] = 1;
}

// ---------------------------------------------------------------------------
// 5) renoise update where (mism & cont): hash-based Box-Muller normal
// ---------------------------------------------------------------------------
__global__ void update_kernel(const float* __restrict__ data,
                              const int* __restrict__ timesteps,
                              float* __restrict__ wsf, int it) {
  const int gid = blockIdx.x * blockDim.x + threadIdx.x;   // 0..BS*NS-1
  if (gid >= BS * NS) return;
  const int* ctr = (const int*)(wsf + OFF_CTR);
  if (!ctr[8 + it]) return;
  const int* mismB = (const int*)(wsf + OFF_MISM);
  if (!mismB[gid]) return;

  const int b = gid >> 10;
  const int t = timesteps[b];
  const float sa = wsf[OFF_SA + t], s1 = wsf[OFF_S1 + t];
  float*       noiseS = wsf + OFF_NOISE + (size_t)gid * DIM;
  float*       noisyS = wsf + OFF_NOISY + (size_t)gid * DIM;
  const float* dp     = data + (size_t)gid * DIM;

  #pragma unroll
  for (int d = 0; d < DIM; ++d) {
    const unsigned s  = (unsigned)(gid * DIM + d) ^ (0x9E3779B9u * (unsigned)(it + 1));
    const unsigned h1 = hash_u32(s ^ 0x85ebca6bu);
    const unsigned h2 = hash_u32(s ^ 0xc2b2ae35u);
    const float u1 = ((float)h1 + 1.0f) * 2.3283064365386963e-10f;  // (0,1]
    const float u2 = (float)h2 * 2.3283064365386963e-10f;
    const float z  = sqrtf(-2.0f * logf(u1)) * cosf(6.28318530717958647f * u2);
    noiseS[d] = z;
    noisyS[d] = sa * dp[d] + s1 * z;
  }
}

// ---------------------------------------------------------------------------
// 6) zero output
// ---------------------------------------------------------------------------
__global__ void zero_out_kernel(float* __restrict__ out) {
  if (threadIdx.x < BS) out[threadIdx.x] = 0.f;
}

// ---------------------------------------------------------------------------
// 7) WMMA conditional-MLP + epsilon MSE loss over free dims {3,4,5}.
//    grid = BS*NS/16 blocks x 128 threads (4 waves per 16-token tile).
//    GEMM1: A = x tile (16x6, K padded to 8), B = W1 chunks, C = bias.
//    GEMM2: A = tanh(h) from LDS (K=512 split over 4 waves), B = W2 (N pad 16).
// ---------------------------------------------------------------------------
__global__ void mlp_loss_kernel(const float* __restrict__ data,
                                const float* __restrict__ W1,
                                const float* __restrict__ W2,
                                const float* __restrict__ b2,
                                const float* __restrict__ wsf,
                                float* __restrict__ out) {
  constexpr int HSTRIDE = HID + 8;   // 520 (even), LDS row stride
  const int tile = blockIdx.x;                 // BS*NS/16 = 8192
  const int b    = tile >> 6;
  const int n0   = (tile & 63) << 4;
  const int tid  = threadIdx.x;                // 128
  const int wave = tid >> 5;
  const int lane = tid & 31;
  const bool hi  = lane >= 16;
  const int  lm  = lane & 15;

  __shared__ __align__(32) float hbuf[16 * HSTRIDE];  // tanh activations
  __shared__ float part[4][16 * 16];                  // GEMM2 partials per wave
  __shared__ float ssum;

  const float* bias   = wsf + OFF_BIAS  + (size_t)b * HID;
  const float* noiseS = wsf + OFF_NOISE;
  const float* noisyS = wsf + OFF_NOISY;

  __builtin_prefetch(W2 + tid * 16, 0, 1);     // global_prefetch_b8

  // A operand: noisy_final row = {data[0..2], noisy[3..5]} (built once)
  const int tok = n0 + lm;
  const float* dp = data   + ((size_t)b * NS + tok) * DIM;
  const float* np = noisyS + ((size_t)b * NS + tok) * DIM;
  v2f a0, a1;
  a0[0] = hi ? dp[2] : dp[0];
  a0[1] = hi ? np[3] : dp[1];
  a1[0] = hi ? 0.f   : np[4];
  a1[1] = hi ? 0.f   : np[5];

  // ---- GEMM1: each wave produces 8 of the 32 hid chunks ----
  const int rbase = hi ? 8 : 0;
  const int r0 = hi ? 2 : 0;        // W1 rows feeding w0 pair
  const int r1 = hi ? 3 : 1;
  for (int j = 0; j < 8; ++j) {
    const int hcol = (wave * 8 + j) * 16 + lm;
    // unconditional loads + branchless zero padding
    v2f w0, w1v;
    w0[0] = W1[r0 * HID + hcol];
    w0[1] = W1[r1 * HID + hcol];
    const float w4 = W1[4 * HID + hcol];
    const float w5 = W1[5 * HID + hcol];
    w1v[0] = hi ? 0.f : w4;
    w1v[1] = hi ? 0.f : w5;
    const float bb = bias[hcol];
    v8f c;
    #pragma unroll
    for (int v = 0; v < 8; ++v) c[v] = bb;
    c = wmma4(a0, w0, c);
    c = wmma4(a1, w1v, c);
    #pragma unroll
    for (int v = 0; v < 8; ++v) hbuf[(rbase + v) * HSTRIDE + hcol] = tanhf(c[v]);
  }
  if (tid == 0) ssum = 0.f;
  __syncthreads();

  // ---- GEMM2: wave accumulates K range [wave*128, wave*128+128) ----
  const int   ncol  = lm;                      // output dim, valid 0..5
  const int   ncolc = (ncol < DIM) ? ncol : 0; // clamped -> loads always legal
  const float wm    = (ncol < DIM) ? 1.f : 0.f;
  v8f c2;
  #pragma unroll
  for (int v = 0; v < 8; ++v) c2[v] = 0.f;
  for (int kk = 0; kk < 128; kk += 4) {
    const int kA = wave * 128 + kk + (hi ? 2 : 0);   // even -> 8B aligned
    const float2 pa = *(const float2*)&hbuf[lm * HSTRIDE + kA];
    v2f a, w;
    a[0] = pa.x; a[1] = pa.y;
    w[0] = W2[kA * DIM + ncolc] * wm;
    w[1] = W2[(kA + 1) * DIM + ncolc] * wm;
    c2 = wmma4(a, w, c2);
  }
  #pragma unroll
  for (int v = 0; v < 8; ++v) part[wave][(rbase + v) * 16 + ncol] = c2[v];
  __syncthreads();

  // ---- loss on free dims 3..5 ----
  if (tid < 48) {
    const int tk = tid / 3, d = 3 + tid % 3;
    float p = part[0][tk * 16 + d] + part[1][tk * 16 + d]
            + part[2][tk * 16 + d] + part[3][tk * 16 + d] + b2[d];
    const float tg = noiseS[((size_t)b * NS + n0 + tk) * DIM + d];
    const float diff = p - tg;
    atomicAdd(&ssum, diff * diff);
  }
  __syncthreads();
  if (tid == 0) atomicAdd(&out[b], ssum * (1.0f / ((float)NS * 3.0f)));
}

// ---------------------------------------------------------------------------
// launch
// ---------------------------------------------------------------------------
extern "C" void kernel_launch(void* const* d_in, const int* in_sizes, int n_in,
                              void* d_out, int out_size, void* d_ws, size_t ws_size,
                              hipStream_t stream) {
  const float* data      = (const float*)d_in[0];
  const float* context   = (const float*)d_in[1];
  const float* noise0    = (const float*)d_in[2];
  const float* W1        = (const float*)d_in[3];
  const float* Wc        = (const float*)d_in[4];
  const float* Wt        = (const float*)d_in[5];
  const float* b1        = (const float*)d_in[6];
  const float* W2        = (const float*)d_in[7];
  const float* b2        = (const float*)d_in[8];
  const int*   timesteps = (const int*)d_in[9];
  float* out = (float*)d_out;
  float* wsf = (float*)d_ws;

  tables_kernel<<<1, 32, 0, stream>>>(wsf);

  const int nelem = BS * NS * DIM;
  prep_kernel<<<(nelem + 255) / 256, 256, 0, stream>>>(data, noise0, timesteps, wsf);

  bias_kernel<<<BS, 256, 0, stream>>>(context, Wc, Wt, b1, timesteps, wsf);

  for (int it = 0; it < ITERS; ++it) {
    argmin_kernel<<<BS * 8, 256, 0, stream>>>(data, wsf, it);
    decide_kernel<<<1, 32, 0, stream>>>(wsf, it);
    update_kernel<<<(BS * NS + 255) / 256, 256, 0, stream>>>(data, timesteps, wsf, it);
  }

  zero_out_kernel<<<1, 128, 0, stream>>>(out);
  mlp_loss_kernel<<<(BS * NS) / 16, 128, 0, stream>>>(data, W1, W2, b2, wsf, out);
}